// LGCNLayer_69561290326178
// MI455X (gfx1250) — compile-verified
//
#include <hip/hip_runtime.h>
#include <hip/hip_bf16.h>

typedef __attribute__((ext_vector_type(2))) float v2f;
typedef __attribute__((ext_vector_type(8))) float v8f;

#define D 64
#define SCAN_T 256
#define SCAN_ELEMS 4
#define SCAN_CHUNK (SCAN_T * SCAN_ELEMS)   // 1024 elements per scan block

// ---------------------------------------------------------------------------
// WMMA fp32 GEMM:  Out[M x 64] = A[M x 64] @ W[64 x 64]
// One wave per 16x16 output tile, 16 chained V_WMMA_F32_16X16X4_F32.
// ---------------------------------------------------------------------------
__global__ __launch_bounds__(128)
void gemm_wmma_f32(const float* __restrict__ A, const float* __restrict__ W,
                   float* __restrict__ Out, int M) {
    const int lane  = threadIdx.x;          // 0..31
    const int tileM = blockIdx.x * 16;
    const int tileN = threadIdx.y * 16;     // 0,16,32,48
    const int half  = lane >> 4;
    const int m     = lane & 15;

    int rowA = tileM + m;
    if (rowA >= M) rowA = M - 1;
    const float* arow = A + (long)rowA * D;

    v8f c = {};
#pragma unroll
    for (int ks = 0; ks < D; ks += 4) {
        v2f afrag = *(const v2f*)(arow + ks + 2 * half);
        v2f bfrag;
        bfrag.x = W[(ks + 2 * half + 0) * D + tileN + m];
        bfrag.y = W[(ks + 2 * half + 1) * D + tileN + m];
        c = __builtin_amdgcn_wmma_f32_16x16x4_f32(false, afrag, false, bfrag,
                                                  (short)0, c, false, false);
    }

#pragma unroll
    for (int r = 0; r < 8; ++r) {
        int row = tileM + half * 8 + r;
        if (row < M) Out[(long)row * D + tileN + m] = c[r];
    }
}

// ---------------------------------------------------------------------------
// Per-node attention halves: a[i]=feat[i].lin_w[0:64], b[i]=feat[i].lin_w[64:128]
// ---------------------------------------------------------------------------
__global__ void attn_pre_kernel(const float* __restrict__ feat,
                                const float* __restrict__ lin_w,
                                float* __restrict__ a, float* __restrict__ b, int N) {
    int i = blockIdx.x * blockDim.x + threadIdx.x;
    if (i >= N) return;
    const float* f = feat + (long)i * D;
    float av = 0.f, bv = 0.f;
#pragma unroll 8
    for (int k = 0; k < D; ++k) {
        float fv = f[k];
        av += fv * lin_w[k];
        bv += fv * lin_w[D + k];
    }
    a[i] = av;
    b[i] = bv;
}

// ---------------------------------------------------------------------------
// CSR build phase 1: in-degree histogram
// ---------------------------------------------------------------------------
__global__ void hist_kernel(const int* __restrict__ edst,
                            unsigned* __restrict__ count, int E) {
    int e = blockIdx.x * blockDim.x + threadIdx.x;
    if (e < E) atomicAdd(&count[edst[e]], 1u);
}

// ---------------------------------------------------------------------------
// CSR build phase 2: device-wide exclusive scan (block scan + 49 partials)
// ---------------------------------------------------------------------------
__global__ __launch_bounds__(SCAN_T)
void scan_block_kernel(const unsigned* __restrict__ count,
                       unsigned* __restrict__ offsets,
                       unsigned* __restrict__ blockSums, int N) {
    __shared__ unsigned sdata[SCAN_T];
    int t = threadIdx.x;
    int base = blockIdx.x * SCAN_CHUNK + t * SCAN_ELEMS;
    unsigned v[SCAN_ELEMS];
    unsigned sum = 0;
#pragma unroll
    for (int i = 0; i < SCAN_ELEMS; ++i) {
        v[i] = (base + i < N) ? count[base + i] : 0u;
        sum += v[i];
    }
    sdata[t] = sum;
    __syncthreads();
    // Hillis-Steele inclusive scan over 256 thread sums
    for (int off = 1; off < SCAN_T; off <<= 1) {
        unsigned x = (t >= off) ? sdata[t - off] : 0u;
        __syncthreads();
        sdata[t] += x;
        __syncthreads();
    }
    unsigned run = sdata[t] - sum;    // exclusive prefix for this thread
#pragma unroll
    for (int i = 0; i < SCAN_ELEMS; ++i) {
        if (base + i < N) offsets[base + i] = run;
        run += v[i];
    }
    if (t == SCAN_T - 1) blockSums[blockIdx.x] = sdata[t];
}

__global__ void scan_tops_kernel(unsigned* __restrict__ blockSums, int nb) {
    if (blockIdx.x == 0 && threadIdx.x == 0) {
        unsigned run = 0;
        for (int i = 0; i < nb; ++i) {
            unsigned x = blockSums[i];
            blockSums[i] = run;
            run += x;
        }
    }
}

__global__ void scan_add_kernel(unsigned* __restrict__ offsets,
                                const unsigned* __restrict__ blockSums, int N) {
    int i = blockIdx.x * blockDim.x + threadIdx.x;
    if (i < N) offsets[i] += blockSums[i / SCAN_CHUNK];
}

// ---------------------------------------------------------------------------
// CSR build phase 3: fill {src, att} slots grouped by destination.
// att = sigmoid(relu(a[src] + b[dst] + bias))
// ---------------------------------------------------------------------------
__global__ void fill_kernel(const int* __restrict__ esrc, const int* __restrict__ edst,
                            const float* __restrict__ a, const float* __restrict__ b,
                            const float* __restrict__ lin_b,
                            const unsigned* __restrict__ offsets,
                            unsigned* __restrict__ cursor,
                            int2* __restrict__ slots, int E) {
    int e = blockIdx.x * blockDim.x + threadIdx.x;
    if (e >= E) return;
    int s = esrc[e];
    int d = edst[e];
    float x   = a[s] + b[d] + lin_b[0];
    float att = 1.0f / (1.0f + __expf(-fmaxf(x, 0.0f)));
    unsigned slot = offsets[d] + atomicAdd(&cursor[d], 1u);
    slots[slot] = make_int2(s, __float_as_int(att));
}

// ---------------------------------------------------------------------------
// Gather + finalize (fused): one wave per destination node, zero fp32 atomics.
// out = tanh( (cnt>0 ? sum(att*P[src]) : feat) * norm + (cnt>0 ? L : EL) )
// ---------------------------------------------------------------------------
__global__ __launch_bounds__(256)
void gather_finalize_kernel(const int2* __restrict__ slots,
                            const unsigned* __restrict__ offsets,
                            const unsigned* __restrict__ count,
                            const float* __restrict__ P,
                            const float* __restrict__ feat,
                            const float* __restrict__ norm,
                            const float* __restrict__ L,
                            const float* __restrict__ EL,
                            float* __restrict__ out, int N) {
    int node = (blockIdx.x * blockDim.x + threadIdx.x) >> 5;
    int lane = threadIdx.x & 31;
    if (node >= N) return;

    unsigned begin = offsets[node];
    unsigned cnt   = count[node];

    float accx = 0.f, accy = 0.f;
    for (unsigned j = 0; j < cnt; ++j) {
        int2 sl   = slots[begin + j];              // wave-uniform load
        float att = __int_as_float(sl.y);
        const float2 p = *(const float2*)(P + (size_t)sl.x * D + lane * 2);
        accx = fmaf(att, p.x, accx);
        accy = fmaf(att, p.y, accy);
    }

    bool  has = cnt > 0u;
    float nrm = norm[node];
    size_t idx = (size_t)node * D + lane * 2;
    float2 f = *(const float2*)(feat + idx);
    float2 l = has ? *(const float2*)(L + idx) : *(const float2*)(EL + idx);
    float bx = has ? accx : f.x;
    float by = has ? accy : f.y;
    float2 o;
    o.x = tanhf(fmaf(bx, nrm, l.x));
    o.y = tanhf(fmaf(by, nrm, l.y));
    *(float2*)(out + idx) = o;
}

// ---------------------------------------------------------------------------
extern "C" void kernel_launch(void* const* d_in, const int* in_sizes, int n_in,
                              void* d_out, int out_size, void* d_ws, size_t ws_size,
                              hipStream_t stream) {
    const float* feat   = (const float*)d_in[0];   // [N, 64]
    const float* norm   = (const float*)d_in[1];   // [N]
    const int*   esrc   = (const int*)  d_in[2];   // [E]
    const int*   edst   = (const int*)  d_in[3];   // [E]
    /* etype d_in[4] unused: relation split is a no-op in the reference */
    const float* W_rel  = (const float*)d_in[5];   // [64, 64]
    const float* lin_w  = (const float*)d_in[6];   // [128, 1]
    const float* lin_b  = (const float*)d_in[7];   // [1]
    const float* loop_w = (const float*)d_in[8];   // [64, 64]
    const float* ev_w   = (const float*)d_in[9];   // [64, 64]

    const int N = in_sizes[0] / D;
    const int E = in_sizes[2];
    const int nb = (N + SCAN_CHUNK - 1) / SCAN_CHUNK;   // scan partials

    float* out = (float*)d_out;

    // Workspace layout (all fp32/u32; slots aligned to 16B at the end)
    float*    P      = (float*)d_ws;                   // [N,64]
    float*    Lw     = P   + (size_t)N * D;            // [N,64]
    float*    ELw    = Lw  + (size_t)N * D;            // [N,64]
    float*    aHalf  = ELw + (size_t)N * D;            // [N]
    float*    bHalf  = aHalf + N;                      // [N]
    unsigned* count  = (unsigned*)(bHalf + N);         // [N]
    unsigned* offs   = count + N;                      // [N]
    unsigned* cursor = offs + N;                       // [N]
    unsigned* bsums  = cursor + N;                     // [nb] (pad to 1024)
    size_t slotOff = ((size_t)(bsums + 1024) - (size_t)d_ws + 15) & ~(size_t)15;
    int2*     slots  = (int2*)((char*)d_ws + slotOff); // [E] {src, att}

    hipMemsetAsync(count,  0, (size_t)N * sizeof(unsigned), stream);
    hipMemsetAsync(cursor, 0, (size_t)N * sizeof(unsigned), stream);

    // Dense node-level GEMMs via fp32 WMMA
    dim3 gblk(32, 4);
    dim3 ggrd((N + 15) / 16);
    gemm_wmma_f32<<<ggrd, gblk, 0, stream>>>(feat, W_rel,  P,   N);
    gemm_wmma_f32<<<ggrd, gblk, 0, stream>>>(feat, loop_w, Lw,  N);
    gemm_wmma_f32<<<ggrd, gblk, 0, stream>>>(feat, ev_w,   ELw, N);

    // Per-node attention halves
    attn_pre_kernel<<<(N + 255) / 256, 256, 0, stream>>>(feat, lin_w, aHalf, bHalf, N);

    // CSR-by-destination build
    hist_kernel<<<(E + 255) / 256, 256, 0, stream>>>(edst, count, E);
    scan_block_kernel<<<nb, SCAN_T, 0, stream>>>(count, offs, bsums, N);
    scan_tops_kernel<<<1, 32, 0, stream>>>(bsums, nb);
    scan_add_kernel<<<(N + 255) / 256, 256, 0, stream>>>(offs, bsums, N);
    fill_kernel<<<(E + 255) / 256, 256, 0, stream>>>(esrc, edst, aHalf, bHalf,
                                                     lin_b, offs, cursor, slots, E);

    // Fused gather + finalize: one wave per node, register accumulation
    {
        long threads = (long)N * 32;
        int blocks = (int)((threads + 255) / 256);
        gather_finalize_kernel<<<blocks, 256, 0, stream>>>(slots, offs, count, P,
                                                           feat, norm, Lw, ELw, out, N);
    }
}